// GeographicalCrossEntropyLoss_38104949850242
// MI455X (gfx1250) — compile-verified
//
#include <hip/hip_runtime.h>
#include <cmath>
#include <stdint.h>

// ---------------- problem constants ----------------
#define B_ROWS   262144
#define C_CLS    195
#define EARTH_R  6371.0f

// ---------------- tiling ----------------
#define THREADS     128            // 4 waves of 32
#define WAVES       4
#define TILE_ROWS   64             // 4 waves * 16 rows
#define TILE_ELEMS  (TILE_ROWS * C_CLS)   // 12480 floats = 49920 B
#define TILE_VEC4   (TILE_ELEMS / 4)      // 3120 x 16B (49920 % 16 == 0)
#define NBLOCKS     (B_ROWS / TILE_ROWS)  // 4096

typedef float v2f __attribute__((ext_vector_type(2)));
typedef float v8f __attribute__((ext_vector_type(8)));
typedef __attribute__((address_space(3))) float lds_f32;

__global__ __launch_bounds__(THREADS)
void geo_ce_main(const float* __restrict__ logits,
                 const int*   __restrict__ target,
                 const float* __restrict__ coords,
                 float*       __restrict__ partial) {
    __shared__ float tile[TILE_ELEMS];     // 64 rows x 195 cols, flat (== global layout)
    __shared__ float cshare[C_CLS * 2];    // lat/lon pairs
    __shared__ float wsums[WAVES][16];     // per-wave row sums from WMMA
    __shared__ float wpart[WAVES];         // per-wave partial contribution

    const int tid  = threadIdx.x;
    const int lane = tid & 31;
    const int wave = tid >> 5;

    // ---- stage coords into LDS (tiny, plain loads) ----
    for (int i = tid; i < C_CLS * 2; i += THREADS) cshare[i] = coords[i];

    // ---- stage 64x195 logits tile into LDS via async global->LDS b128 (ASYNCcnt path) ----
    const long long elemBase = (long long)blockIdx.x * TILE_ELEMS;   // 16B-aligned region
    {
        const uint64_t gbase = (uint64_t)(uintptr_t)(logits + elemBase);    // wave-uniform
        const uint32_t tbase = (uint32_t)(uintptr_t)(lds_f32*)&tile[0];     // LDS byte offset
        for (int i = tid; i < TILE_VEC4; i += THREADS) {
            const uint32_t ldsb = tbase + (uint32_t)i * 16u;
            const uint32_t goff = (uint32_t)i * 16u;
            asm volatile("global_load_async_to_lds_b128 %0, %1, %2"
                         :: "v"(ldsb), "v"(goff), "s"(gbase) : "memory");
        }
        asm volatile("s_wait_asynccnt 0" ::: "memory");   // this wave's transfers done
    }
    __syncthreads();                                      // tile visible to all waves

    // ---- each wave owns rows [wave*16, wave*16+16); lane pair (l, l+16) per row ----
    const int r        = lane & 15;            // row within 16-row group
    const int half     = lane >> 4;            // 0 or 1
    const int rowLocal = wave * 16 + r;
    const float* rowp  = &tile[rowLocal * C_CLS];

    // pass 1: max + argmax. half 0: cols [0,98), half 1: cols [97,195) (uniform trip count)
    float m  = -__builtin_inff();
    int   am = 0;
    {
        const int c0 = half * 97;
        #pragma unroll 4
        for (int k = 0; k < 98; ++k) {
            const int c = c0 + k;
            const float x = rowp[c];
            if (x > m) { m = x; am = c; }
        }
        // symmetric pair-combine; ties prefer lower column index (argmax-first rule)
        const float mo  = __shfl_xor(m, 16);
        const int   amo = __shfl_xor(am, 16);
        if (mo > m || (mo == m && amo < am)) { m = mo; am = amo; }
    }

    // pass 2: sum(exp(x - m)) for 16 rows at once via V_WMMA_F32_16X16X4_F32.
    // A layout (32-bit A 16x4): lanes 0-15 hold row M=lane, K={0,1}; lanes 16-31 K={2,3}.
    // B = ones  =>  D[M][N] = sum_K A[M][K]; accumulate the f32 C chain over K-chunks.
    v8f acc = {0.f, 0.f, 0.f, 0.f, 0.f, 0.f, 0.f, 0.f};
    {
        const v2f bones = {1.0f, 1.0f};
        // 48 full chunks, completely branch-free (cols 0..191)
        #pragma unroll 4
        for (int cc = 0; cc < 192; cc += 4) {
            const int ca = cc + half * 2;
            const v2f a = { __expf(rowp[ca]     - m),
                            __expf(rowp[ca + 1] - m) };
            acc = __builtin_amdgcn_wmma_f32_16x16x4_f32(
                false, a, false, bones, (short)0, acc, false, false);
        }
        // epilogue chunk: cols 192..194; the 4th element (col 195) exists only as a
        // select-to-zero (no EXEC divergence): half==1 lanes re-read col 194 and mask.
        {
            const int ca = 192 + half * 2;              // 192 or 194
            const float e0 = __expf(rowp[ca]            - m);
            const float e1 = __expf(rowp[ca + 1 - half] - m);   // 193 or 194 (valid)
            const v2f a = { e0, half ? 0.0f : e1 };
            acc = __builtin_amdgcn_wmma_f32_16x16x4_f32(
                false, a, false, bones, (short)0, acc, false, false);
        }
    }

    // Extract per-row sums: C/D layout -> lane 0 vgpr j = (M=j, N=0); lane 16 vgpr j = (M=j+8, N=0)
    if (lane == 0 || lane == 16) {
        #pragma unroll
        for (int j = 0; j < 8; ++j) wsums[wave][half * 8 + j] = acc[j];
    }
    // same-wave DS ordering: wait for the stores, then read
    asm volatile("s_wait_dscnt 0" ::: "memory");

    // ---- lanes 0..15 finalize one row each ----
    float contrib = 0.0f;
    if (lane < 16) {
        const float s = wsums[wave][lane];
        const long long grow = (long long)blockIdx.x * TILE_ROWS + rowLocal;
        const int t = target[grow];

        // cross entropy term: -(x_t - m - log(sum))
        const float ce = -(rowp[t] - m - __logf(s));

        // haversine(coords[argmax], coords[target])
        const float D2R = 0.017453292519943295f;
        const float lat1 = cshare[2 * am]     * D2R;
        const float lon1 = cshare[2 * am + 1] * D2R;
        const float lat2 = cshare[2 * t]      * D2R;
        const float lon2 = cshare[2 * t + 1]  * D2R;
        const float sdlat = __sinf(0.5f * (lat2 - lat1));
        const float sdlon = __sinf(0.5f * (lon2 - lon1));
        float ha = sdlat * sdlat + __cosf(lat1) * __cosf(lat2) * sdlon * sdlon;
        ha = fminf(fmaxf(ha, 0.0f), 1.0f);
        const float dist = 2.0f * EARTH_R * asinf(sqrtf(ha));

        contrib = ce * (1.0f / (float)B_ROWS)
                + dist * (1.0f / (500.0f * (float)B_ROWS));
    }

    // deterministic fixed-pattern wave reduction
    #pragma unroll
    for (int off = 16; off > 0; off >>= 1) contrib += __shfl_xor(contrib, off);
    if (lane == 0) wpart[wave] = contrib;
    __syncthreads();
    if (tid == 0) partial[blockIdx.x] = wpart[0] + wpart[1] + wpart[2] + wpart[3];
}

__global__ __launch_bounds__(256)
void geo_ce_reduce(const float* __restrict__ partial, int n, float* __restrict__ out) {
    __shared__ float sm[256];
    float s = 0.0f;
    for (int i = threadIdx.x; i < n; i += 256) s += partial[i];   // fixed order
    sm[threadIdx.x] = s;
    __syncthreads();
    #pragma unroll
    for (int st = 128; st > 0; st >>= 1) {
        if (threadIdx.x < st) sm[threadIdx.x] += sm[threadIdx.x + st];
        __syncthreads();
    }
    if (threadIdx.x == 0) out[0] = sm[0];
}

extern "C" void kernel_launch(void* const* d_in, const int* in_sizes, int n_in,
                              void* d_out, int out_size, void* d_ws, size_t ws_size,
                              hipStream_t stream) {
    const float* logits = (const float*)d_in[0];   // [B, C] f32
    const int*   target = (const int*)d_in[1];     // [B] i32
    const float* coords = (const float*)d_in[2];   // [C, 2] f32
    float* out     = (float*)d_out;                // scalar f32
    float* partial = (float*)d_ws;                 // NBLOCKS floats (16 KB)

    geo_ce_main<<<NBLOCKS, THREADS, 0, stream>>>(logits, target, coords, partial);
    geo_ce_reduce<<<1, 256, 0, stream>>>(partial, NBLOCKS, out);
    (void)in_sizes; (void)n_in; (void)out_size; (void)ws_size;
}